// STGCNSpatialConv_17282948399325
// MI455X (gfx1250) — compile-verified
//
#include <hip/hip_runtime.h>
#include <hip/hip_bf16.h>

// ---------------------------------------------------------------------------
// STGCN spatial conv for MI455X (gfx1250, wave32, WMMA).
// FP32 matrix work mapped onto V_WMMA_F32_16X16X4_F32.
// Algebraic folds: W_fus fusion collapsed into Wcomb = iw*Wg@Wf_top and a
// per-row vector through Wcomb2 = gw*Wg@Wf_bot, so key_intra / the 128-wide
// fusion GEMM are never materialized.
// ---------------------------------------------------------------------------

typedef float v2f __attribute__((ext_vector_type(2)));
typedef float v8f __attribute__((ext_vector_type(8)));

__device__ __forceinline__ v8f wmma4(v2f a, v2f b, v8f c) {
  // D = A(16x4) * B(4x16) + C(16x16), fp32
  return __builtin_amdgcn_wmma_f32_16x16x4_f32(
      /*neg_a=*/false, a, /*neg_b=*/false, b,
      /*c_mod=*/(short)0, c, /*reuse_a=*/false, /*reuse_b=*/false);
}

constexpr int Bb = 8, Ss = 64, Nn = 2048, Ff = 64;
constexpr int Kk = 256, Cc = 128, Mm = 14;
constexpr float EPS = 1e-5f;

// ws layout (floats): [0,4096) Wcomb ; [4096,8192) Wcomb2 ; [8192,8256) cbase
// ---------------------------------------------------------------------------
__global__ void k0_precompute(const float* __restrict__ Wg,
                              const float* __restrict__ bg,
                              const float* __restrict__ Wfus,
                              const float* __restrict__ bfus,
                              const float* __restrict__ intra_w,
                              const float* __restrict__ global_w,
                              float* __restrict__ ws) {
  const float iw = intra_w[0], gw = global_w[0];
  float* Wcomb  = ws;
  float* Wcomb2 = ws + 4096;
  float* cbase  = ws + 8192;
  int t = threadIdx.x;
  for (int o = t; o < 4096; o += 256) {
    int j = o >> 6, f = o & 63;
    float s1 = 0.f, s2 = 0.f;
    for (int k = 0; k < 64; ++k) {
      float w = Wg[j * 64 + k];
      s1 += w * Wfus[k * 64 + f];
      s2 += w * Wfus[(64 + k) * 64 + f];
    }
    Wcomb[o]  = iw * s1;
    Wcomb2[o] = gw * s2;
  }
  if (t < 64) {
    float s = bfus[t];
    for (int k = 0; k < 64; ++k)
      s += iw * bg[k] * Wfus[k * 64 + t] + gw * bg[k] * Wfus[(64 + k) * 64 + t];
    cbase[t] = s;
  }
}

// ---------------------------------------------------------------------------
// Key branch: one workgroup (8 waves) per (b,s).
// P = key_A @ LN(key_X)  -> accumulators (8 16x16 tiles per wave)
// out = P @ Wcomb + cvec, cvec from column-mean of P via Wg / LN / Wcomb2.
// ---------------------------------------------------------------------------
__global__ __launch_bounds__(256, 1) void k_key(
    const float* __restrict__ X, const float* __restrict__ adj,
    const float* __restrict__ Wg, const float* __restrict__ bg,
    const int* __restrict__ key_idx, const float* __restrict__ ws,
    float* __restrict__ out) {
  __shared__ float lnb[64][66];   // LN'ed key_X chunk (B operand)
  __shared__ float pst[64][66];   // P staging for second GEMM
  __shared__ float colsum[64];
  __shared__ float mvec[64], krow[64], lnk[64], cvec[64];
  __shared__ int kidx[256];

  const int bs = blockIdx.x;
  const int b = bs >> 6, s = bs & 63;
  const float* Xbs = X + (size_t)(b * Ss + s) * Nn * Ff;
  float* outbs = out + (size_t)(b * Ss + s) * Nn * Ff;

  const int tid = threadIdx.x;
  const int wave = tid >> 5, lane = tid & 31;
  const int ln16 = lane & 15, hi = lane >> 4;

  for (int i = tid; i < 256; i += 256) kidx[i] = key_idx[i];
  __syncthreads();

  const v8f vzero = {0.f, 0.f, 0.f, 0.f, 0.f, 0.f, 0.f, 0.f};
  v8f P[2][4];
#pragma unroll
  for (int i = 0; i < 2; ++i)
#pragma unroll
    for (int j = 0; j < 4; ++j) P[i][j] = vzero;

  // ---- GEMM1: P = key_A @ LN(key_X), K streamed in 4 chunks of 64 ----
  for (int ch = 0; ch < 4; ++ch) {
    // LN 64 rows into LDS: 4 threads per row, 16 elems each
    {
      const int r = tid >> 2, q = tid & 3;
      const float* xr = Xbs + (size_t)kidx[ch * 64 + r] * Ff;
      float v[16];
      float sum = 0.f, sq = 0.f;
#pragma unroll
      for (int e = 0; e < 16; ++e) {
        float x = xr[q * 16 + e];
        v[e] = x; sum += x; sq += x * x;
      }
      sum += __shfl_xor(sum, 1); sum += __shfl_xor(sum, 2);
      sq  += __shfl_xor(sq, 1);  sq  += __shfl_xor(sq, 2);
      const float mu = sum * (1.f / 64.f);
      const float var = sq * (1.f / 64.f) - mu * mu;
      const float rstd = rsqrtf(var + EPS);
#pragma unroll
      for (int e = 0; e < 16; ++e) lnb[r][q * 16 + e] = (v[e] - mu) * rstd;
    }
    __syncthreads();

    for (int k0 = 0; k0 < 64; k0 += 4) {
      const int ka = ch * 64 + k0 + 2 * hi;
      const int ac0 = kidx[ka], ac1 = kidx[ka + 1];
      v2f bf[4];
#pragma unroll
      for (int j = 0; j < 4; ++j) {
        bf[j].x = lnb[k0 + 2 * hi][j * 16 + ln16];
        bf[j].y = lnb[k0 + 2 * hi + 1][j * 16 + ln16];
      }
#pragma unroll
      for (int i = 0; i < 2; ++i) {
        const int grow = (2 * wave + i) * 16 + ln16;
        const float* arow = adj + (size_t)kidx[grow] * Nn;
        v2f af; af.x = arow[ac0]; af.y = arow[ac1];
#pragma unroll
        for (int j = 0; j < 4; ++j) P[i][j] = wmma4(af, bf[j], P[i][j]);
      }
    }
    __syncthreads();
  }

  // ---- column mean of P (over all 256 rows) ----
  if (tid < 64) colsum[tid] = 0.f;
  __syncthreads();
#pragma unroll
  for (int i = 0; i < 2; ++i)
#pragma unroll
    for (int j = 0; j < 4; ++j) {
      float p = 0.f;
#pragma unroll
      for (int c = 0; c < 8; ++c) p += P[i][j][c];
      p += __shfl_xor(p, 16);
      if (hi == 0) atomicAdd(&colsum[j * 16 + ln16], p);
    }
  __syncthreads();

  // ---- tiny vector chain: mean -> @Wg+bg -> LN -> @Wcomb2 + cbase ----
  if (tid < 64) mvec[tid] = colsum[tid] * (1.f / 256.f);
  __syncthreads();
  if (tid < 64) {
    float a = bg[tid];
    for (int j = 0; j < 64; ++j) a += mvec[j] * Wg[j * 64 + tid];
    krow[tid] = a;
  }
  __syncthreads();
  if (tid < 64) {
    float mu = 0.f, sq = 0.f;
    for (int j = 0; j < 64; ++j) { float x = krow[j]; mu += x; sq += x * x; }
    mu *= (1.f / 64.f);
    const float var = sq * (1.f / 64.f) - mu * mu;
    lnk[tid] = (krow[tid] - mu) * rsqrtf(var + EPS);
  }
  __syncthreads();
  if (tid < 64) {
    const float* Wcomb2 = ws + 4096;
    const float* cbase  = ws + 8192;
    float a = cbase[tid];
    for (int j = 0; j < 64; ++j) a += lnk[j] * Wcomb2[j * 64 + tid];
    cvec[tid] = a;
  }
  __syncthreads();

  // ---- GEMM2: out = P @ Wcomb + cvec, per 64-row block via LDS staging ----
  const float* Wcomb = ws;
  for (int blk = 0; blk < 4; ++blk) {
    if ((wave >> 1) == blk) {
#pragma unroll
      for (int i = 0; i < 2; ++i) {
        const int tm = 2 * wave + i;
        const int trb = tm - blk * 4;
#pragma unroll
        for (int j = 0; j < 4; ++j)
#pragma unroll
          for (int c = 0; c < 8; ++c)
            pst[trb * 16 + c + 8 * hi][j * 16 + ln16] = P[i][j][c];
      }
    }
    __syncthreads();
#pragma unroll
    for (int tt = 0; tt < 2; ++tt) {
      const int t = 2 * wave + tt;
      const int tm2 = t >> 2, tn2 = t & 3;
      const float cv = cvec[tn2 * 16 + ln16];
      v8f acc = {cv, cv, cv, cv, cv, cv, cv, cv};
      for (int k0 = 0; k0 < 64; k0 += 4) {
        v2f a;
        a.x = pst[tm2 * 16 + ln16][k0 + 2 * hi];
        a.y = pst[tm2 * 16 + ln16][k0 + 2 * hi + 1];
        v2f bb;
        bb.x = Wcomb[(k0 + 2 * hi) * 64 + tn2 * 16 + ln16];
        bb.y = Wcomb[(k0 + 2 * hi + 1) * 64 + tn2 * 16 + ln16];
        acc = wmma4(a, bb, acc);
      }
      const int gm0 = blk * 64 + tm2 * 16;
#pragma unroll
      for (int c = 0; c < 8; ++c) {
        const int gr = kidx[gm0 + c + 8 * hi];
        outbs[(size_t)gr * 64 + tn2 * 16 + ln16] = acc[c];
      }
    }
    __syncthreads();
  }
}

// ---------------------------------------------------------------------------
// Cluster branch: one 64-thread (2-wave) workgroup per (b,s,c).
// ---------------------------------------------------------------------------
__global__ __launch_bounds__(64, 1) void k_cluster(
    const float* __restrict__ X, const float* __restrict__ adj,
    const float* __restrict__ Wg, const float* __restrict__ bg,
    const float* __restrict__ W_imp, const float* __restrict__ b_imp,
    const int* __restrict__ cluster_idx, const float* __restrict__ ws,
    float* __restrict__ out) {
  __shared__ float cxs[16][66];
  __shared__ float lnc[16][66];
  __shared__ float cAt[16][18];
  __shared__ float qst[16][66];
  __shared__ float imp[16];
  __shared__ float cent[64], lncent[64], cvec[64];
  __shared__ int idx[16];

  const int wg = blockIdx.x;
  const int c = wg % Cc;
  const int bs = wg / Cc;
  const int b = bs >> 6, s = bs & 63;
  const float* Xbs = X + (size_t)(b * Ss + s) * Nn * Ff;
  float* outbs = out + (size_t)(b * Ss + s) * Nn * Ff;

  const int t = threadIdx.x;  // 0..63
  const int wave = t >> 5, lane = t & 31;
  const int ln16 = lane & 15, hi = lane >> 4;

  if (t < 16) idx[t] = (t < Mm) ? cluster_idx[c * Mm + t] : 0;
  __syncthreads();

  // load 14x64 tile (zero-padded to 16 rows), column = t
  for (int m = 0; m < 16; ++m)
    cxs[m][t] = (m < Mm) ? Xbs[(size_t)idx[m] * Ff + t] : 0.f;
  __syncthreads();

  // importance logits + softmax over M
  if (t < 16) {
    float lg = b_imp[0];
    if (t < Mm) for (int f = 0; f < 64; ++f) lg += cxs[t][f] * W_imp[f];
    imp[t] = lg;
  }
  __syncthreads();
  float myimp = 0.f;
  if (t < 16) {
    float mx = -1e30f;
    for (int m = 0; m < Mm; ++m) mx = fmaxf(mx, imp[m]);
    float den = 0.f;
    for (int m = 0; m < Mm; ++m) den += __expf(imp[m] - mx);
    myimp = (t < Mm) ? __expf(imp[t] - mx) / den : 0.f;
  }
  __syncthreads();
  if (t < 16) imp[t] = myimp;
  __syncthreads();

  // centers[f]
  {
    float a = 0.f;
    for (int m = 0; m < Mm; ++m) a += cxs[m][t] * imp[m];
    cent[t] = a;
  }
  // row LN: 4 threads per row for rows 0..13; pad rows 14,15 with zeros
  if (t < 56) {
    const int r = t >> 2, q = t & 3;
    float sum = 0.f, sq = 0.f;
    float v[16];
#pragma unroll
    for (int e = 0; e < 16; ++e) {
      float x = cxs[r][q * 16 + e];
      v[e] = x; sum += x; sq += x * x;
    }
    sum += __shfl_xor(sum, 1); sum += __shfl_xor(sum, 2);
    sq  += __shfl_xor(sq, 1);  sq  += __shfl_xor(sq, 2);
    const float mu = sum * (1.f / 64.f);
    const float var = sq * (1.f / 64.f) - mu * mu;
    const float rstd = rsqrtf(var + EPS);
#pragma unroll
    for (int e = 0; e < 16; ++e) lnc[r][q * 16 + e] = (v[e] - mu) * rstd;
  } else {
    const int t2 = t - 56;  // 0..7 -> zero rows 14,15
    const int r = 14 + (t2 >> 2), q = t2 & 3;
#pragma unroll
    for (int e = 0; e < 16; ++e) lnc[r][q * 16 + e] = 0.f;
  }
  // cA (zero-padded 16x16)
  for (int o = t; o < 256; o += 64) {
    const int n = o >> 4, m = o & 15;
    cAt[n][m] = (n < Mm && m < Mm) ? adj[(size_t)idx[n] * Nn + idx[m]] : 0.f;
  }
  __syncthreads();

  // LN(centers) -> cvec = cbase + lncent @ Wcomb2
  {
    float mu = 0.f, sq = 0.f;
    for (int j = 0; j < 64; ++j) { float x = cent[j]; mu += x; sq += x * x; }
    mu *= (1.f / 64.f);
    const float var = sq * (1.f / 64.f) - mu * mu;
    lncent[t] = (cent[t] - mu) * rsqrtf(var + EPS);
  }
  __syncthreads();
  {
    const float* Wcomb2 = ws + 4096;
    const float* cbase  = ws + 8192;
    float a = cbase[t];
    for (int j = 0; j < 64; ++j) a += lncent[j] * Wcomb2[j * 64 + t];
    cvec[t] = a;
  }

  // GEMM1: Q(16x64) = cA(16x16) @ lnc(16x64); 2 tiles per wave
  const v8f vzero = {0.f, 0.f, 0.f, 0.f, 0.f, 0.f, 0.f, 0.f};
  v8f q[2] = {vzero, vzero};
  for (int k0 = 0; k0 < 16; k0 += 4) {
    v2f a;
    a.x = cAt[ln16][k0 + 2 * hi];
    a.y = cAt[ln16][k0 + 2 * hi + 1];
#pragma unroll
    for (int tt = 0; tt < 2; ++tt) {
      const int tn = 2 * wave + tt;
      v2f bb;
      bb.x = lnc[k0 + 2 * hi][tn * 16 + ln16];
      bb.y = lnc[k0 + 2 * hi + 1][tn * 16 + ln16];
      q[tt] = wmma4(a, bb, q[tt]);
    }
  }
#pragma unroll
  for (int tt = 0; tt < 2; ++tt) {
    const int tn = 2 * wave + tt;
#pragma unroll
    for (int cc = 0; cc < 8; ++cc)
      qst[cc + 8 * hi][tn * 16 + ln16] = q[tt][cc];
  }
  __syncthreads();

  // GEMM2: out(16x64) = Q @ Wcomb + cvec; store rows < 14
  const float* Wcomb = ws;
#pragma unroll
  for (int tt = 0; tt < 2; ++tt) {
    const int tn = 2 * wave + tt;
    const float cv = cvec[tn * 16 + ln16];
    v8f acc = {cv, cv, cv, cv, cv, cv, cv, cv};
    for (int k0 = 0; k0 < 64; k0 += 4) {
      v2f a;
      a.x = qst[ln16][k0 + 2 * hi];
      a.y = qst[ln16][k0 + 2 * hi + 1];
      v2f bb;
      bb.x = Wcomb[(k0 + 2 * hi) * 64 + tn * 16 + ln16];
      bb.y = Wcomb[(k0 + 2 * hi + 1) * 64 + tn * 16 + ln16];
      acc = wmma4(a, bb, acc);
    }
#pragma unroll
    for (int cc = 0; cc < 8; ++cc) {
      const int m = cc + 8 * hi;
      if (m < Mm) outbs[(size_t)idx[m] * Ff + tn * 16 + ln16] = acc[cc];
    }
  }
}

// ---------------------------------------------------------------------------
extern "C" void kernel_launch(void* const* d_in, const int* in_sizes, int n_in,
                              void* d_out, int out_size, void* d_ws,
                              size_t ws_size, hipStream_t stream) {
  const float* X        = (const float*)d_in[0];
  const float* adj      = (const float*)d_in[1];
  const float* Wg       = (const float*)d_in[2];
  const float* bg       = (const float*)d_in[3];
  const float* W_imp    = (const float*)d_in[4];
  const float* b_imp    = (const float*)d_in[5];
  const float* W_fus    = (const float*)d_in[6];
  const float* b_fus    = (const float*)d_in[7];
  const float* intra_w  = (const float*)d_in[8];
  const float* global_w = (const float*)d_in[9];
  const int* key_idx    = (const int*)d_in[10];
  const int* cluster_i  = (const int*)d_in[11];
  float* out = (float*)d_out;
  float* ws  = (float*)d_ws;

  k0_precompute<<<1, 256, 0, stream>>>(Wg, bg, W_fus, b_fus, intra_w, global_w, ws);
  k_key<<<Bb * Ss, 256, 0, stream>>>(X, adj, Wg, bg, key_idx, ws, out);
  k_cluster<<<Bb * Ss * Cc, 64, 0, stream>>>(X, adj, Wg, bg, W_imp, b_imp,
                                             cluster_i, ws, out);
}